// SSM_node_89017492177040
// MI455X (gfx1250) — compile-verified
//
#include <hip/hip_runtime.h>

typedef float v2f __attribute__((ext_vector_type(2)));
typedef float v4f __attribute__((ext_vector_type(4)));
typedef float v8f __attribute__((ext_vector_type(8)));

#define NEG_BIG (-3.0e38f)

// ---------------------------------------------------------------------------
// Kernel A: HMM prior recurrence in probability domain using WMMA f32 16x16x4.
//   p_t = p_{t-1} * P   (P = row-softmax of transition), h_t = log(p_t).
// One wave32. A-matrix = p replicated over all 16 rows (so every D row equals
// p_next); B = P split into 4 K=4 chunks; 4 chained WMMAs per step.
// ---------------------------------------------------------------------------
__global__ __launch_bounds__(32) void hmm_recurrence_wmma(
    const float* __restrict__ init_raw,   // [16]
    const float* __restrict__ trans_raw,  // [16,16]
    float* __restrict__ h_out,            // [T,16]
    int T)
{
    __shared__ float Pm[16][16];
    const int lane = threadIdx.x;
    const int half = lane >> 4;     // 0 | 1
    const int idx  = lane & 15;

    // Row-wise softmax of transition -> probability matrix in LDS
    if (lane < 16) {
        float row[16];
        float m = NEG_BIG;
#pragma unroll
        for (int j = 0; j < 16; ++j) {
            row[j] = trans_raw[lane * 16 + j];
            m = fmaxf(m, row[j]);
        }
        float ssum = 0.0f;
#pragma unroll
        for (int j = 0; j < 16; ++j) ssum += expf(row[j] - m);
        const float lse = m + logf(ssum);
#pragma unroll
        for (int j = 0; j < 16; ++j) Pm[lane][j] = expf(row[j] - lse);
    }
    __syncthreads();

    // log_softmax of state init (computed redundantly on every lane)
    float xs[16];
    float m0 = NEG_BIG;
#pragma unroll
    for (int i = 0; i < 16; ++i) {
        xs[i] = init_raw[i];
        m0 = fmaxf(m0, xs[i]);
    }
    float s0 = 0.0f;
#pragma unroll
    for (int i = 0; i < 16; ++i) s0 += expf(xs[i] - m0);
    const float l0 = m0 + logf(s0);
    const float h0 = xs[idx] - l0;          // lane L: h0[L&15]
    if (lane < 16) h_out[lane] = h0;        // t = 0 row
    float pn = expf(h0);                    // lane L: p[L&15]

    // B chunks: B[c] vgpr0 = Pm[4c + 2*half][idx], vgpr1 = next K row
    v2f B[4];
#pragma unroll
    for (int c = 0; c < 4; ++c) {
        const int k0 = 4 * c + 2 * half;
        B[c].x = Pm[k0][idx];
        B[c].y = Pm[k0 + 1][idx];
    }

    for (int t = 1; t < T; ++t) {
        // Build replicated-A chunks from pn via intra-wave shuffles
        v2f A[4];
#pragma unroll
        for (int c = 0; c < 4; ++c) {
            const int k0 = 4 * c + 2 * half;      // always < 16
            A[c].x = __shfl(pn, k0, 32);
            A[c].y = __shfl(pn, k0 + 1, 32);
        }
        v8f acc = {};
#pragma unroll
        for (int c = 0; c < 4; ++c) {
            acc = __builtin_amdgcn_wmma_f32_16x16x4_f32(
                false, A[c], false, B[c], (short)0, acc, false, false);
        }
        pn = acc[0];                 // lane L: p_next[L&15] (all D rows equal)
        const float hn = logf(pn);
        if (lane < 16) h_out[t * 16 + lane] = hn;
    }
}

// ---------------------------------------------------------------------------
// Kernel B1: per-state LSE over the 2000 nodes of the emission matrix rows.
// ---------------------------------------------------------------------------
__global__ __launch_bounds__(256) void erow_lse(
    const float* __restrict__ e_raw, float* __restrict__ e_lse, int N)
{
    __shared__ float red[256];
    const int s = blockIdx.x;
    const int tid = threadIdx.x;
    const float* row = e_raw + (size_t)s * N;

    float m = NEG_BIG;
    for (int n = tid; n < N; n += 256) m = fmaxf(m, row[n]);
    red[tid] = m;
    __syncthreads();
    for (int off = 128; off > 0; off >>= 1) {
        if (tid < off) red[tid] = fmaxf(red[tid], red[tid + off]);
        __syncthreads();
    }
    m = red[0];
    __syncthreads();

    float ssum = 0.0f;
    for (int n = tid; n < N; n += 256) ssum += expf(row[n] - m);
    red[tid] = ssum;
    __syncthreads();
    for (int off = 128; off > 0; off >>= 1) {
        if (tid < off) red[tid] += red[tid + off];
        __syncthreads();
    }
    if (tid == 0) e_lse[s] = m + logf(red[0]);
}

// ---------------------------------------------------------------------------
// Kernel B2: per-node chain-weight log-softmax; emit transposed wT[c][n] and
// lw_lse[n] = LSE_c of the (rounded) normalized weights (tracks reference).
// ---------------------------------------------------------------------------
__global__ __launch_bounds__(256) void chain_norm(
    const float* __restrict__ w_raw, float* __restrict__ lw_lse,
    float* __restrict__ wT, int N)
{
    const int n = blockIdx.x * 256 + threadIdx.x;
    if (n >= N) return;

    float w[16];
    float m = NEG_BIG;
#pragma unroll
    for (int c = 0; c < 16; ++c) {
        w[c] = w_raw[n * 16 + c];
        m = fmaxf(m, w[c]);
    }
    float ssum = 0.0f;
#pragma unroll
    for (int c = 0; c < 16; ++c) ssum += expf(w[c] - m);
    const float lse = m + logf(ssum);

    float wn[16];
    float m2 = NEG_BIG;
#pragma unroll
    for (int c = 0; c < 16; ++c) {
        wn[c] = w[c] - lse;
        m2 = fmaxf(m2, wn[c]);
    }
    float s2 = 0.0f;
#pragma unroll
    for (int c = 0; c < 16; ++c) s2 += expf(wn[c] - m2);
    lw_lse[n] = m2 + logf(s2);

#pragma unroll
    for (int c = 0; c < 16; ++c) wT[(size_t)c * N + n] = wn[c];  // coalesced per c
}

// ---------------------------------------------------------------------------
// Kernel C: the 409.6 MB stream-out  out2[t,s,c,n] = (e[s,n]-e_lse[s]) +
// wT[c,n] + h[t,s].  One block per (t,s); float4 loads + nontemporal stores.
// ---------------------------------------------------------------------------
__global__ __launch_bounds__(256) void big_store(
    const float* __restrict__ e_raw, const float* __restrict__ e_lse,
    const float* __restrict__ wT, const float* __restrict__ h,
    float* __restrict__ out2, int N)
{
    const int ts = blockIdx.x;          // t*16 + s
    const int s  = ts & 15;
    const float bias = h[ts] - e_lse[s];

    const int n4 = N >> 2;              // 500 float4 per row
    const v4f* eRow = (const v4f*)(e_raw + (size_t)s * N);
    v4f* outTS = (v4f*)(out2 + (size_t)ts * 16 * N);

    for (int i = threadIdx.x; i < n4; i += 256) {
        const v4f a = eRow[i] + bias;
#pragma unroll
        for (int c = 0; c < 16; ++c) {
            const v4f w4 = ((const v4f*)(wT + (size_t)c * N))[i];
            const v4f v = a + w4;
            __builtin_nontemporal_store(v, &outTS[(size_t)c * n4 + i]);
        }
    }
}

// ---------------------------------------------------------------------------
// Kernel D: out1[t,n] = LSE_s(e_norm[s,n] + h[t,s]) + lw_lse[n]
// ---------------------------------------------------------------------------
__global__ __launch_bounds__(256) void node_marginal(
    const float* __restrict__ e_raw, const float* __restrict__ e_lse,
    const float* __restrict__ lw_lse, const float* __restrict__ h,
    float* __restrict__ out1, int N)
{
    const int n = blockIdx.x * 256 + threadIdx.x;
    const int t = blockIdx.y;
    if (n >= N) return;

    float v[16];
    float m = NEG_BIG;
#pragma unroll
    for (int s = 0; s < 16; ++s) {
        v[s] = e_raw[(size_t)s * N + n] - e_lse[s] + h[t * 16 + s];
        m = fmaxf(m, v[s]);
    }
    float ssum = 0.0f;
#pragma unroll
    for (int s = 0; s < 16; ++s) ssum += expf(v[s] - m);
    out1[(size_t)t * N + n] = m + logf(ssum) + lw_lse[n];
}

// ---------------------------------------------------------------------------
extern "C" void kernel_launch(void* const* d_in, const int* in_sizes, int n_in,
                              void* d_out, int out_size, void* d_ws, size_t ws_size,
                              hipStream_t stream)
{
    const float* init_raw = (const float*)d_in[0];  // [16]
    const float* w_raw    = (const float*)d_in[1];  // [N,16]
    const float* e_raw    = (const float*)d_in[2];  // [16,N]
    const float* t_raw    = (const float*)d_in[3];  // [16,16]
    // d_in[4] = num_iters lives on device; derive T from output size instead.

    const int N = in_sizes[2] / 16;                         // 2000
    const int T = out_size / (N + 16 * 16 * N + 16);        // 200

    float* out1 = (float*)d_out;                            // [T,N]
    float* out2 = out1 + (size_t)T * N;                     // [T,16,16,N]
    float* hptr = out2 + (size_t)T * 16 * 16 * N;           // [T,16]

    float* e_lse  = (float*)d_ws;       // 16
    float* lw_lse = e_lse + 16;         // N
    float* wT     = lw_lse + N;         // 16*N   (~136 KB total)

    hmm_recurrence_wmma<<<1, 32, 0, stream>>>(init_raw, t_raw, hptr, T);
    erow_lse<<<16, 256, 0, stream>>>(e_raw, e_lse, N);
    chain_norm<<<(N + 255) / 256, 256, 0, stream>>>(w_raw, lw_lse, wT, N);
    big_store<<<T * 16, 256, 0, stream>>>(e_raw, e_lse, wT, hptr, out2, N);
    dim3 gD((N + 255) / 256, T);
    node_marginal<<<gD, 256, 0, stream>>>(e_raw, e_lse, lw_lse, hptr, out1, N);
}